// PPISPSlangPy_38886633898212
// MI455X (gfx1250) — compile-verified
//
#include <hip/hip_runtime.h>
#include <stdint.h>

#define THREADS 256
#define BLOCKS  2048

// Native clang vectors (HIP_vector_type structs are rejected by the
// __builtin_nontemporal_* builtins; ext_vector_type works and still
// lowers to single b128 VMEM ops).
typedef __attribute__((ext_vector_type(4))) float vf4;
typedef __attribute__((ext_vector_type(4))) int   vi4;

// ---------- fast native transcendentals (v_exp_f32 / v_log_f32 / v_rcp_f32) ----------
__device__ __forceinline__ float fexp2(float x) { return __builtin_amdgcn_exp2f(x); }
__device__ __forceinline__ float flog2(float x) { return __builtin_amdgcn_logf(x); }
__device__ __forceinline__ float frcp (float x) { return __builtin_amdgcn_rcpf(x); }
__device__ __forceinline__ float fast_exp(float x) { return fexp2(x * 1.4426950408889634f); }
// softplus(x) = ln(1 + e^x) = ln2 * log2(1 + e^x)
__device__ __forceinline__ float softplus_f(float x) {
    return 0.6931471805599453f * flog2(1.0f + fast_exp(x));
}

// ---------- CDNA5 async global->LDS fill (ASYNCcnt path, gfx1250-only) ----------
// lds_byte_addr: logical LDS byte address (low 32 bits of the flat shared pointer).
// Each active lane copies 16B per issued instruction; whole block strides the table.
__device__ __forceinline__ void async_fill_b128(unsigned lds_byte_addr,
                                                const void* gsrc, int nbytes, int tid) {
    const char* g = (const char*)gsrc;
    for (int off = tid * 16; off < nbytes; off += THREADS * 16) {
        unsigned la = lds_byte_addr + (unsigned)off;
        const void* ga = (const void*)(g + off);
        asm volatile("global_load_async_to_lds_b128 %0, %1, off"
                     :: "v"(la), "v"(ga)
                     : "memory");
    }
}

// ---------- per-element ISP (tables already pre-transformed in LDS) ----------
// sExpo[f]        = exp(exposure[f])
// sCol[f*8+0/1]   = exp(cp0/cp1), rest raw
// sCrf[cam*12+ch*4 + {0,1,2}] = 0.3+sp(k1), 0.3+sp(k2), 0.1+sp(k3); +3 = k4 raw
__device__ __forceinline__ void ppisp_one(
    const float* __restrict__ sExpo, const float* __restrict__ sVig,
    const float* __restrict__ sCol,  const float* __restrict__ sCrf,
    float inw, float inh,
    float r_in, float g_in, float b_in,
    float pxx, float pyy, int ci, int fi,
    float* __restrict__ o) {
    const float ex = sExpo[fi];
    float xs[3] = { r_in * ex, g_in * ex, b_in * ex };

    const float nx = pxx * inw - 0.5f;
    const float ny = pyy * inh - 0.5f;
    const float* vp = sVig + ci * 15;
#pragma unroll
    for (int ch = 0; ch < 3; ++ch) {
        const float dx = nx - vp[ch * 5 + 0];
        const float dy = ny - vp[ch * 5 + 1];
        const float r2 = dx * dx + dy * dy;
        const float vg = 1.0f + r2 * (vp[ch * 5 + 2] + r2 * (vp[ch * 5 + 3] + r2 * vp[ch * 5 + 4]));
        xs[ch] *= vg;
    }

    const vf4 ca = *(const vf4*)(sCol + fi * 8);
    const vf4 cb = *(const vf4*)(sCol + fi * 8 + 4);
    const float rr = xs[0] * ca.x;   // ca.x = exp(cp0)
    const float gg = xs[1];
    const float bb = xs[2] * ca.y;   // ca.y = exp(cp1)
    float y[3];
    y[0] = rr + ca.z * gg + ca.w * bb;
    y[1] = cb.x * rr + gg + cb.y * bb;
    y[2] = cb.z * rr + cb.w * gg + bb;

#pragma unroll
    for (int ch = 0; ch < 3; ++ch) {
        const vf4 kp = *(const vf4*)(sCrf + ci * 12 + ch * 4);
        const float xg = fexp2(kp.z * flog2(fmaxf(y[ch], 1e-6f)));   // x^k3
        o[ch] = kp.x * xg * frcp(fmaf(kp.x - 1.0f, xg, kp.y)) + kp.w;
    }
}

__global__ __launch_bounds__(THREADS) void ppisp_kernel(
    const float* __restrict__ rgb, const float* __restrict__ pix,
    const int*   __restrict__ cam, const int*   __restrict__ frm,
    const float* __restrict__ gExpo, const float* __restrict__ gVig,
    const float* __restrict__ gCol,  const float* __restrict__ gCrf,
    float* __restrict__ out,
    int groups, int rem,
    int nExpo, int nVig, int nCol, int nCrf,
    const int* __restrict__ resw, const int* __restrict__ resh) {
    // Layout: [exposure | vignetting | color | crf] -- 1000 + 120 + 8000 + 96 floats = 36864 B.
    // All section byte sizes/offsets are multiples of 16 for b128 async copies.
    __shared__ __align__(16) float lds[9216];
    const int tid = threadIdx.x;

    float* sExpo = lds;
    float* sVig  = sExpo + nExpo;
    float* sCol  = sVig + nVig;
    float* sCrf  = sCol + nCol;

    const unsigned lbase = (unsigned)(unsigned long long)(uintptr_t)&lds[0];
    async_fill_b128(lbase,                                   gExpo, nExpo * 4, tid);
    async_fill_b128(lbase + (unsigned)(nExpo * 4),           gVig,  nVig  * 4, tid);
    async_fill_b128(lbase + (unsigned)((nExpo + nVig) * 4),  gCol,  nCol  * 4, tid);
    async_fill_b128(lbase + (unsigned)((nExpo + nVig + nCol) * 4), gCrf, nCrf * 4, tid);
    asm volatile("s_wait_asynccnt 0" ::: "memory");
    __syncthreads();

    // One-time table transforms: kills 21 of 27 per-element transcendentals.
    for (int i = tid; i < nExpo; i += THREADS) sExpo[i] = fast_exp(sExpo[i]);
    const int nFrames = nCol >> 3;
    for (int i = tid; i < nFrames; i += THREADS) {
        sCol[i * 8 + 0] = fast_exp(sCol[i * 8 + 0]);
        sCol[i * 8 + 1] = fast_exp(sCol[i * 8 + 1]);
    }
    const int nCrfRows = nCrf >> 2;   // cameras * 3 channels
    for (int i = tid; i < nCrfRows; i += THREADS) {
        float* kp = sCrf + i * 4;
        kp[0] = 0.3f + softplus_f(kp[0]);
        kp[1] = 0.3f + softplus_f(kp[1]);
        kp[2] = 0.1f + softplus_f(kp[2]);
    }
    __syncthreads();

    const float inw = frcp((float)resw[0]);
    const float inh = frcp((float)resh[0]);

    const vf4* rgb4 = (const vf4*)rgb;
    const vf4* pix4 = (const vf4*)pix;
    const vi4* cam4 = (const vi4*)cam;
    const vi4* frm4 = (const vi4*)frm;
    vf4*       out4 = (vf4*)out;

    const int stride = (int)gridDim.x * THREADS;
    for (int g = (int)blockIdx.x * THREADS + tid; g < groups; g += stride) {
        // 4 elements per thread: all streaming traffic is b128 with non-temporal
        // hints (single-use stream; keep the 192MB L2 for the hot tables).
        const vf4 t0 = __builtin_nontemporal_load(rgb4 + 3 * g + 0);
        const vf4 t1 = __builtin_nontemporal_load(rgb4 + 3 * g + 1);
        const vf4 t2 = __builtin_nontemporal_load(rgb4 + 3 * g + 2);
        const vf4 p0 = __builtin_nontemporal_load(pix4 + 2 * g + 0);
        const vf4 p1 = __builtin_nontemporal_load(pix4 + 2 * g + 1);
        const vi4 c4 = __builtin_nontemporal_load(cam4 + g);
        const vi4 f4 = __builtin_nontemporal_load(frm4 + g);

        float o[12];
        ppisp_one(sExpo, sVig, sCol, sCrf, inw, inh, t0.x, t0.y, t0.z, p0.x, p0.y, c4.x, f4.x, o + 0);
        ppisp_one(sExpo, sVig, sCol, sCrf, inw, inh, t0.w, t1.x, t1.y, p0.z, p0.w, c4.y, f4.y, o + 3);
        ppisp_one(sExpo, sVig, sCol, sCrf, inw, inh, t1.z, t1.w, t2.x, p1.x, p1.y, c4.z, f4.z, o + 6);
        ppisp_one(sExpo, sVig, sCol, sCrf, inw, inh, t2.y, t2.z, t2.w, p1.z, p1.w, c4.w, f4.w, o + 9);

        const vf4 s0 = { o[0], o[1], o[2],  o[3]  };
        const vf4 s1 = { o[4], o[5], o[6],  o[7]  };
        const vf4 s2 = { o[8], o[9], o[10], o[11] };
        __builtin_nontemporal_store(s0, out4 + 3 * g + 0);
        __builtin_nontemporal_store(s1, out4 + 3 * g + 1);
        __builtin_nontemporal_store(s2, out4 + 3 * g + 2);
    }

    // Tail (N % 4 != 0): scalar path in block 0.
    if (blockIdx.x == 0 && tid < rem) {
        const int e = groups * 4 + tid;
        float o[3];
        ppisp_one(sExpo, sVig, sCol, sCrf, inw, inh,
                  rgb[e * 3 + 0], rgb[e * 3 + 1], rgb[e * 3 + 2],
                  pix[e * 2 + 0], pix[e * 2 + 1], cam[e], frm[e], o);
        out[e * 3 + 0] = o[0];
        out[e * 3 + 1] = o[1];
        out[e * 3 + 2] = o[2];
    }
}

extern "C" void kernel_launch(void* const* d_in, const int* in_sizes, int n_in,
                              void* d_out, int out_size, void* d_ws, size_t ws_size,
                              hipStream_t stream) {
    (void)n_in; (void)out_size; (void)d_ws; (void)ws_size;
    const float* rgb  = (const float*)d_in[0];
    const float* pix  = (const float*)d_in[1];
    const int*   cam  = (const int*)d_in[2];
    const int*   frm  = (const int*)d_in[3];
    const float* expo = (const float*)d_in[4];
    const float* vig  = (const float*)d_in[5];
    const float* col  = (const float*)d_in[6];
    const float* crf  = (const float*)d_in[7];
    const int*   resw = (const int*)d_in[8];
    const int*   resh = (const int*)d_in[9];
    float* out = (float*)d_out;

    const int N      = in_sizes[0] / 3;
    const int groups = N >> 2;
    const int rem    = N - (groups << 2);

    int blocks = BLOCKS;
    const int needed = (groups + THREADS - 1) / THREADS;
    if (needed < blocks) blocks = (needed > 0) ? needed : 1;

    ppisp_kernel<<<blocks, THREADS, 0, stream>>>(
        rgb, pix, cam, frm, expo, vig, col, crf, out,
        groups, rem,
        in_sizes[4], in_sizes[5], in_sizes[6], in_sizes[7],
        resw, resh);
}